// AttentionLoss_85761906966695
// MI455X (gfx1250) — compile-verified
//
#include <hip/hip_runtime.h>

// Guided-attention loss, MI455X (gfx1250).
// Bandwidth-bound reduction; TDM (tensor_load_to_lds) streams only the valid
// [ml x tl] rectangle of each batch into LDS (OOB -> zero-fill == free masking),
// double-buffered with s_wait_tensorcnt; wave32 shuffle reductions.

#define GDIAG   0.2f
#define BATCH   64
#define MEL     2000
#define TEXTN   256
#define TR      16                    // tile rows per TDM transfer (16KB tiles)
#define BAND    128                   // rows per workgroup
#define NTILES  (BAND / TR)           // 8
#define GY      ((MEL + BAND - 1) / BAND)   // 16
#define NPART   (BATCH * GY)          // 1024 partials

typedef unsigned int v4u __attribute__((ext_vector_type(4)));
typedef int          v4i __attribute__((ext_vector_type(4)));
typedef int          v8i __attribute__((ext_vector_type(8)));

// Build a 2D Tensor-DMA descriptor (D# groups 0/1) and issue TENSOR_LOAD_TO_LDS.
// tile = TR rows x 256 cols of f32 starting at gaddr; reads are clamped to
// tensor_dim0 = tdim0 columns and tensor_dim1 = tdim1 rows (zero-filled beyond).
__device__ __forceinline__ void tdm_load_tile(unsigned lds_off,
                                              unsigned long long gaddr,
                                              unsigned tdim0, unsigned tdim1)
{
    v4u g0;
    g0.x = 1u;                                           // count=1, user desc
    g0.y = lds_off;                                      // LDS byte address
    g0.z = (unsigned)(gaddr & 0xFFFFFFFFull);            // global_addr[31:0]
    g0.w = (unsigned)((gaddr >> 32) & 0x1FFFFFFull)      // global_addr[56:32]
         | 0x80000000u;                                  // type=2 ("image")

    v8i g1;
    g1[0] = (int)(2u << 16);                             // data_size=2 -> 4 bytes
    g1[1] = (int)((tdim0 & 0xFFFFu) << 16);              // tensor_dim0[15:0]
    g1[2] = (int)((tdim0 >> 16) | ((tdim1 & 0xFFFFu) << 16));   // dim0 hi | dim1 lo
    g1[3] = (int)((tdim1 >> 16) | ((unsigned)TEXTN << 16));     // dim1 hi | tile_dim0=256
    g1[4] = (int)TR;                                     // tile_dim1=TR, tile_dim2=0
    g1[5] = TEXTN;                                       // tensor_dim0_stride = 256 elems
    g1[6] = 0;
    g1[7] = 0;

    v4i z4 = {0, 0, 0, 0};
#if defined(__clang_major__) && (__clang_major__ >= 23)
    v8i z8 = {0, 0, 0, 0, 0, 0, 0, 0};
    __builtin_amdgcn_tensor_load_to_lds(g0, g1, z4, z4, z8, 0);
#else
    __builtin_amdgcn_tensor_load_to_lds(g0, g1, z4, z4, 0);
#endif
}

__global__ void __launch_bounds__(256)
guided_attn_partial(const float* __restrict__ pred,
                    const int* __restrict__ text_len,
                    const int* __restrict__ mel_len,
                    float* __restrict__ partial)
{
    __shared__ float smem[2][TR * TEXTN];   // 2 x 16KB double buffer
    __shared__ float wsum[8];

    const int b    = blockIdx.x;
    const int r0   = blockIdx.y * BAND;
    const int tid  = threadIdx.x;
    const int lane = tid & 31;
    const int wid  = tid >> 5;

    const int tl = __builtin_amdgcn_readfirstlane(text_len[b]);
    const int ml = __builtin_amdgcn_readfirstlane(mel_len[b]);

    float acc = 0.0f;
    const int rows = min(BAND, min(MEL - r0, ml - r0));   // block-uniform
    if (rows > 0) {
        const int   ntiles = (rows + TR - 1) / TR;
        const float inv_tl = 1.0f / (float)tl;
        const float inv_ml = 1.0f / (float)ml;
        const float C      = -1.0f / (2.0f * GDIAG * GDIAG);   // -12.5

        const unsigned long long base =
            (unsigned long long)(uintptr_t)pred +
            ((unsigned long long)b * MEL * TEXTN +
             (unsigned long long)r0 * TEXTN) * 4ull;
        const unsigned lds0 = (unsigned)(uintptr_t)&smem[0][0]; // low 32b = LDS offset
        const unsigned lds1 = (unsigned)(uintptr_t)&smem[1][0];

        if (wid == 0) {                               // wave-uniform: TDM ignores EXEC
            tdm_load_tile(lds0, base, (unsigned)tl, (unsigned)(ml - r0));
        }
        for (int k = 0; k < ntiles; ++k) {
            if (wid == 0) {
                if (k + 1 < ntiles) {                 // prefetch next tile
                    tdm_load_tile((k & 1) ? lds0 : lds1,
                                  base + (unsigned long long)(k + 1) * TR * TEXTN * 4ull,
                                  (unsigned)tl,
                                  (unsigned)(ml - r0 - (k + 1) * TR));
                    __builtin_amdgcn_s_wait_tensorcnt(1);   // tile k done (in-order)
                } else {
                    __builtin_amdgcn_s_wait_tensorcnt(0);
                }
            }
            __syncthreads();

            const float* buf = (k & 1) ? &smem[1][0] : &smem[0][0];
            // 8 waves x (TR/8)=2 rows each; lanes sweep 256 cols as 2x float4
            #pragma unroll
            for (int rr = 0; rr < TR / 8; ++rr) {
                const int   row = wid * (TR / 8) + rr;
                const int   t   = r0 + k * TR + row;
                const float tml = (float)t * inv_ml;
                #pragma unroll
                for (int h = 0; h < 2; ++h) {
                    const int    c0 = h * 128 + lane * 4;
                    const float4 p  = *(const float4*)&buf[row * TEXTN + c0];
                    float x = (float)c0 * inv_tl - tml;   // n/tl - t/ml, incremental in n
                    #pragma unroll
                    for (int j = 0; j < 4; ++j) {
                        const float e  = __expf(x * x * C);      // v_exp_f32
                        const float pv = (&p.x)[j];              // pv==0 outside valid
                        acc += pv - e * pv;                      // (1-e)*pred
                        x += inv_tl;
                    }
                }
            }
            __syncthreads();
        }
    }

    // wave32 reduction, then cross-wave via LDS
    for (int off = 16; off > 0; off >>= 1)
        acc += __shfl_down(acc, off, 32);
    if (lane == 0) wsum[wid] = acc;
    __syncthreads();
    if (tid == 0) {
        float s = 0.0f;
        #pragma unroll
        for (int w = 0; w < 8; ++w) s += wsum[w];
        partial[blockIdx.y * BATCH + b] = s;
    }
}

__global__ void __launch_bounds__(256)
guided_attn_finalize(const float* __restrict__ partial,
                     const int* __restrict__ text_len,
                     const int* __restrict__ mel_len,
                     float* __restrict__ out)
{
    __shared__ float wsum[8];
    const int tid = threadIdx.x;
    float s = 0.0f;
    for (int i = tid; i < NPART; i += 256) s += partial[i];
    for (int off = 16; off > 0; off >>= 1)
        s += __shfl_down(s, off, 32);
    if ((tid & 31) == 0) wsum[tid >> 5] = s;
    __syncthreads();
    if (tid == 0) {
        float tot = 0.0f;
        #pragma unroll
        for (int w = 0; w < 8; ++w) tot += wsum[w];
        float active = 0.0f;
        for (int i = 0; i < BATCH; ++i)
            active += (float)text_len[i] * (float)mel_len[i];
        out[0] = tot / active * 1.0f;   // ATTN_WEIGHT = 1.0
    }
}

extern "C" void kernel_launch(void* const* d_in, const int* in_sizes, int n_in,
                              void* d_out, int out_size, void* d_ws, size_t ws_size,
                              hipStream_t stream)
{
    // inputs: 0=targets (unused zeros), 1=predictions f32, 2=text_lengths i32, 3=mel_lengths i32
    const float* pred = (const float*)d_in[1];
    const int*   tlen = (const int*)d_in[2];
    const int*   mlen = (const int*)d_in[3];
    float* partial = (float*)d_ws;      // NPART floats of scratch

    dim3 grid(BATCH, GY);
    guided_attn_partial<<<grid, 256, 0, stream>>>(pred, tlen, mlen, partial);
    guided_attn_finalize<<<1, 256, 0, stream>>>(partial, tlen, mlen, (float*)d_out);
}